// ChebyshevEncoder_14319420965217
// MI455X (gfx1250) — compile-verified
//
#include <hip/hip_runtime.h>

// Chebyshev encoder for MI455X (gfx1250).
// Store-bandwidth-bound: 4 MB in, 512 MB out -> ~22 us roofline at 23.3 TB/s.
// Strategy: per-thread serial recurrence into an LDS tile (padded rows),
// then stream the tile to global memory with the CDNA5 async store engine
// (GLOBAL_STORE_ASYNC_FROM_LDS_B128, ASYNCcnt), non-temporal, x4 unrolled,
// for perfectly coalesced 512B-per-wave store bursts.

#define CHEB_ORDER    128
#define TILE_ROWS     64
#define LDS_STRIDE    (CHEB_ORDER + 4)   // floats; 528B rows: 16B aligned, min (2-way) bank conflicts
#define BLOCK_THREADS 64

__global__ __launch_bounds__(BLOCK_THREADS)
void cheb_encode_kernel(const float* __restrict__ x, float* __restrict__ out, int rows) {
    __shared__ __align__(16) float tile[TILE_ROWS * LDS_STRIDE];

    const int tid  = threadIdx.x;
    const int row0 = blockIdx.x * TILE_ROWS;
    const int row  = row0 + tid;

    // ---- Phase 1: serial Chebyshev recurrence, one row per thread, into LDS ----
    if (row < rows) {
        const float xv = x[row];                    // coalesced b32 load
        float* lrow = &tile[tid * LDS_STRIDE];
        float tm2 = 1.0f;        // T0
        float tm1 = xv;          // T1
        float4 c;
        c.x = tm2;
        c.y = tm1;
        {
            float t = 2.0f * xv * tm1 - tm2; tm2 = tm1; tm1 = t; c.z = t;  // T2
            t       = 2.0f * xv * tm1 - tm2; tm2 = tm1; tm1 = t; c.w = t;  // T3
        }
        *reinterpret_cast<float4*>(lrow) = c;       // ds_store_b128
#pragma unroll
        for (int n = 4; n < CHEB_ORDER; n += 4) {
            float t;
            t = 2.0f * xv * tm1 - tm2; tm2 = tm1; tm1 = t; c.x = t;
            t = 2.0f * xv * tm1 - tm2; tm2 = tm1; tm1 = t; c.y = t;
            t = 2.0f * xv * tm1 - tm2; tm2 = tm1; tm1 = t; c.z = t;
            t = 2.0f * xv * tm1 - tm2; tm2 = tm1; tm1 = t; c.w = t;
            *reinterpret_cast<float4*>(lrow + n) = c;
        }
    }
    __syncthreads();

    // ---- Phase 2: async LDS -> global streaming copy of the tile ----
    // Tile is contiguous in global memory: rows row0..row0+vr-1, 128 floats each.
    // Per-thread chunk stride: global +1024B (64 chunks * 16B),
    // LDS +1056B (2 padded rows * 528B) -> constant deltas, so unroll x4.
    const int vr      = min(TILE_ROWS, rows - row0);
    const int nchunks = vr * (CHEB_ORDER / 4);      // 16B chunks; multiple of 32
    const uint64_t gbase =
        (uint64_t)(uintptr_t)out + ((uint64_t)row0 * CHEB_ORDER) * 4ull;

    // Generic->LDS offset: hardware uses addr[31:0] as the LDS byte address.
    const uint32_t lds_base = (uint32_t)(uintptr_t)(&tile[0]);

    int k = tid;
    const int kend4 = nchunks & ~(BLOCK_THREADS * 4 - 1);
    for (; k < kend4; k += BLOCK_THREADS * 4) {
        const int r  = k >> 5;            // element 4k -> row (4k / 128)
        const int cb = (k & 31) << 4;     // byte offset within row (4k % 128)*4
        const uint32_t l0 = lds_base + (uint32_t)(r * (LDS_STRIDE * 4) + cb);
        const uint64_t g0 = gbase + ((uint64_t)k << 4);
        asm volatile("global_store_async_from_lds_b128 %0, %1, off th:TH_STORE_NT"
                     :: "v"(g0), "v"(l0) : "memory");
        asm volatile("global_store_async_from_lds_b128 %0, %1, off th:TH_STORE_NT"
                     :: "v"(g0 + 1024ull), "v"(l0 + 1056u) : "memory");
        asm volatile("global_store_async_from_lds_b128 %0, %1, off th:TH_STORE_NT"
                     :: "v"(g0 + 2048ull), "v"(l0 + 2112u) : "memory");
        asm volatile("global_store_async_from_lds_b128 %0, %1, off th:TH_STORE_NT"
                     :: "v"(g0 + 3072ull), "v"(l0 + 3168u) : "memory");
    }
    for (; k < nchunks; k += BLOCK_THREADS) {
        const int r  = k >> 5;
        const int cb = (k & 31) << 4;
        const uint32_t l0 = lds_base + (uint32_t)(r * (LDS_STRIDE * 4) + cb);
        const uint64_t g0 = gbase + ((uint64_t)k << 4);
        asm volatile("global_store_async_from_lds_b128 %0, %1, off th:TH_STORE_NT"
                     :: "v"(g0), "v"(l0) : "memory");
    }

    // Drain the async store engine before the LDS tile is released.
    asm volatile("s_wait_asynccnt 0x0" ::: "memory");
}

extern "C" void kernel_launch(void* const* d_in, const int* in_sizes, int n_in,
                              void* d_out, int out_size, void* d_ws, size_t ws_size,
                              hipStream_t stream) {
    const float* x = (const float*)d_in[0];
    float* out     = (float*)d_out;
    const int rows = in_sizes[0];                        // N*D = 1,000,000 rows
    const int grid = (rows + TILE_ROWS - 1) / TILE_ROWS; // 15,625 blocks exactly
    hipLaunchKernelGGL(cheb_encode_kernel, dim3(grid), dim3(BLOCK_THREADS), 0, stream,
                       x, out, rows);
}